// ThresholdEncode_83468394430851
// MI455X (gfx1250) — compile-verified
//
#include <hip/hip_runtime.h>
#include <hip/hip_bf16.h>
#include <stdint.h>

// ThresholdEncode for MI455X (gfx1250).
//
// Roofline: 4 MB read + 256 MB write @ 23.3 TB/s HBM => ~11 us floor, pure
// store-bandwidth bound. No GEMM structure -> WMMA not applicable; the win is
// (a) async DMA staging of x into LDS (gfx1250 ASYNCcnt path, kills 16x
// redundant per-lane global reads), (b) coalesced 128-bit non-temporal stores
// (output > L2 capacity and never re-read), (c) wave32 mapping with
// loop-invariant thresholds per lane.

typedef float v4f __attribute__((ext_vector_type(4)));

typedef __attribute__((address_space(1))) int global_int_t;
typedef __attribute__((address_space(3))) int lds_int_t;

#define TPB   256          // 8 waves (wave32)
#define ROWS  512          // rows (pairs) per block
#define ITERS ((ROWS * 16) / TPB)  // float4 stores per thread = 32

__device__ __forceinline__ void async_load_dword_to_lds(const float* gptr, float* lptr) {
#if __has_builtin(__builtin_amdgcn_global_load_async_to_lds_b32)
    __builtin_amdgcn_global_load_async_to_lds_b32(
        (global_int_t*)(uintptr_t)gptr,
        (lds_int_t*)(uintptr_t)lptr,
        /*offset=*/0, /*cpol=*/0);
#else
    unsigned lds_off = (unsigned)(uintptr_t)lptr;             // LDS base is 0 in AS(3)
    unsigned long long gaddr = (unsigned long long)(uintptr_t)gptr;
    asm volatile("global_load_async_to_lds_b32 %0, %1, off"
                 :: "v"(lds_off), "v"(gaddr) : "memory");
#endif
}

__device__ __forceinline__ void wait_asynccnt0() {
#if __has_builtin(__builtin_amdgcn_s_wait_asynccnt)
    __builtin_amdgcn_s_wait_asynccnt(0);
#else
    asm volatile("s_wait_asynccnt 0" ::: "memory");
#endif
}

__global__ __launch_bounds__(TPB)
void ThresholdEncode_83468394430851_kernel(const float* __restrict__ x,
                                           float* __restrict__ out,
                                           int n) {
    __shared__ float sx[ROWS + 1];

    const int t = threadIdx.x;
    const long long r0 = (long long)blockIdx.x * ROWS;

    // ---- Stage x[r0 .. r0+ROWS] (513 dwords) into LDS via async DMA. ----
    // 3 full passes with clamped indices: no EXEC divergence; duplicate lanes
    // write identical data to the same LDS slot (benign).
    #pragma unroll
    for (int p = 0; p < 3; ++p) {
        int k = t + p * TPB;
        if (k > ROWS) k = ROWS;
        long long g = r0 + k;
        if (g > (long long)n - 1) g = (long long)n - 1;
        async_load_dword_to_lds(x + g, &sx[k]);
    }
    wait_asynccnt0();      // this wave's DMA landed in LDS
    __syncthreads();       // all waves' DMA landed

    // ---- Per-lane constants (wave32: lanes 0..15 / 16..31 split rows). ----
    const int   lane16 = t & 15;     // which float4 of the 64-wide row
    const int   rsub   = t >> 4;     // row sub-index within a 16-row stripe
    const float step   = 1.0f / 33.0f;               // matches jnp.linspace delta
    const float th0    = (float)(2 * lane16 + 1) * step;  // threshold j = 2k
    const float th1    = (float)(2 * lane16 + 2) * step;  // threshold j = 2k+1

    v4f* __restrict__ out4 = (v4f*)out;
    const long long obase = r0 * 16;   // float4 units

    // Clamped tail staging makes row n-1 produce xp==xn -> all outputs 0,
    // which is exactly the reference's zero last row. Only rows >= n skipped.
    if (r0 + ROWS <= (long long)n) {
        #pragma unroll
        for (int it = 0; it < ITERS; ++it) {
            const int rl = it * 16 + rsub;
            const float xp = sx[rl];
            const float xn = sx[rl + 1];
            const bool a0 = (xp <= th0), b0 = (xn > th0);
            const bool a1 = (xp <= th1), b1 = (xn > th1);
            v4f o;
            o.x = ( a0 &&  b0) ? 1.0f : 0.0f;   // up_{2k}
            o.y = (!a0 && !b0) ? 1.0f : 0.0f;   // down_{2k}
            o.z = ( a1 &&  b1) ? 1.0f : 0.0f;   // up_{2k+1}
            o.w = (!a1 && !b1) ? 1.0f : 0.0f;   // down_{2k+1}
            __builtin_nontemporal_store(o, &out4[obase + (long long)it * TPB + t]);
        }
    } else {
        #pragma unroll
        for (int it = 0; it < ITERS; ++it) {
            const int rl = it * 16 + rsub;
            if (r0 + rl < (long long)n) {
                const float xp = sx[rl];
                const float xn = sx[rl + 1];
                const bool a0 = (xp <= th0), b0 = (xn > th0);
                const bool a1 = (xp <= th1), b1 = (xn > th1);
                v4f o;
                o.x = ( a0 &&  b0) ? 1.0f : 0.0f;
                o.y = (!a0 && !b0) ? 1.0f : 0.0f;
                o.z = ( a1 &&  b1) ? 1.0f : 0.0f;
                o.w = (!a1 && !b1) ? 1.0f : 0.0f;
                __builtin_nontemporal_store(o, &out4[obase + (long long)it * TPB + t]);
            }
        }
    }
}

extern "C" void kernel_launch(void* const* d_in, const int* in_sizes, int n_in,
                              void* d_out, int out_size, void* d_ws, size_t ws_size,
                              hipStream_t stream) {
    const float* x  = (const float*)d_in[0];
    float* out      = (float*)d_out;
    const int n     = in_sizes[0];          // 1,000,000 samples -> n rows of 64
    const int grid  = (int)(((long long)n + ROWS - 1) / ROWS);
    ThresholdEncode_83468394430851_kernel<<<grid, TPB, 0, stream>>>(x, out, n);
}